// HeteroHANModel_50749333570116
// MI455X (gfx1250) — compile-verified
//
#include <hip/hip_runtime.h>

namespace {

constexpr int kNS  = 100000;
constexpr int kNC  = 10000;
constexpr int kE   = 800000;
constexpr int kHID = 64;
constexpr int kH   = 4;
constexpr int kED  = 32;
constexpr int kHH  = kH * kHID;   // 256
constexpr int kL   = 2;

typedef __attribute__((ext_vector_type(2))) float v2f;
typedef __attribute__((ext_vector_type(8))) float v8f;

__device__ __forceinline__ void atomic_max_f32(float* addr, float v) {
  // monotone int mapping trick; m initialized to -inf bits
  if (v >= 0.0f) {
    atomicMax(reinterpret_cast<int*>(addr), __float_as_int(v));
  } else {
    atomicMin(reinterpret_cast<unsigned int*>(addr),
              static_cast<unsigned int>(__float_as_int(v)));
  }
}

__global__ void fill_kernel(float* __restrict__ p, float v, int n) {
  int i = blockIdx.x * blockDim.x + threadIdx.x;
  if (i < n) p[i] = v;
}

// out[n, c] = emb[idx[n], c]
__global__ void gather_kernel(const float* __restrict__ emb, const int* __restrict__ idx,
                              float* __restrict__ out, int n) {
  int i = blockIdx.x * blockDim.x + threadIdx.x;
  if (i >= n * kHID) return;
  int node = i >> 6, c = i & (kHID - 1);
  out[i] = emb[(size_t)idx[node] * kHID + c];
}

// out[r, h] = sum_c W[r, h*HID + c] * a[h, c]   (collapse W against attention vector)
__global__ void collapse_kernel(const float* __restrict__ W, const float* __restrict__ a,
                                float* __restrict__ out, int rows) {
  int i = blockIdx.x * blockDim.x + threadIdx.x;
  if (i >= rows * kH) return;
  int r = i >> 2, h = i & 3;
  float s = 0.f;
  #pragma unroll
  for (int c = 0; c < kHID; ++c)
    s = fmaf(W[r * kHH + h * kHID + c], a[h * kHID + c], s);
  out[r * kH + h] = s;
}

// C[N,256] = A[N,64] @ W[64,256] via V_WMMA_F32_16X16X4_F32.
// 128 threads = 4 waves; wave w owns cols [64w, 64w+64) as 4 16x16 tiles; block = 16 rows.
__global__ __launch_bounds__(128) void proj_wmma_kernel(
    const float* __restrict__ A, const float* __restrict__ W, float* __restrict__ C) {
  const int lane = threadIdx.x & 31;
  const int wave = threadIdx.x >> 5;
  const int half = lane >> 4;     // 0: K pair {kb,kb+1}, 1: {kb+2,kb+3}
  const int l16  = lane & 15;
  const int row  = blockIdx.x * 16 + l16;
  const int colbase = wave * 64;

  v8f zero = {0.f, 0.f, 0.f, 0.f, 0.f, 0.f, 0.f, 0.f};
  v8f acc[4] = {zero, zero, zero, zero};

  const float* __restrict__ Arow = A + (size_t)row * kHID;
  #pragma unroll
  for (int kb = 0; kb < kHID; kb += 4) {
    const int k0 = kb + half * 2;
    v2f a;
    a.x = Arow[k0];
    a.y = Arow[k0 + 1];
    #pragma unroll
    for (int t = 0; t < 4; ++t) {
      const int col = colbase + t * 16 + l16;
      v2f b;
      b.x = W[k0 * kHH + col];
      b.y = W[(k0 + 1) * kHH + col];
      acc[t] = __builtin_amdgcn_wmma_f32_16x16x4_f32(
          false, a, false, b, (short)0, acc[t], false, false);
    }
  }

  const int mbase = blockIdx.x * 16 + half * 8;   // C layout: VGPR v -> M = v + 8*half
  #pragma unroll
  for (int t = 0; t < 4; ++t) {
    const int col = colbase + t * 16 + l16;
    #pragma unroll
    for (int v = 0; v < 8; ++v)
      C[(size_t)(mbase + v) * kHH + col] = acc[t][v];
  }
}

// alpha[n, h] = x[n, :] @ Wxa[:, h]   (Wxa is [64,4] row-major)
__global__ void node_alpha_kernel(const float* __restrict__ x, const float* __restrict__ Wxa,
                                  float* __restrict__ out, int n) {
  int i = blockIdx.x * blockDim.x + threadIdx.x;
  if (i >= n * kH) return;
  int nd = i >> 2, h = i & 3;
  float s = 0.f;
  #pragma unroll
  for (int k = 0; k < kHID; ++k)
    s = fmaf(x[(size_t)nd * kHID + k], Wxa[k * kH + h], s);
  out[i] = s;
}

// Fused: edge encoder (relu(ea@We+be)) collapsed against Wea, + node alphas, leaky-relu,
// store raw alpha, atomic segment max into m.
__global__ void edge_alpha_kernel(const float* __restrict__ ea,
                                  const float* __restrict__ We, const float* __restrict__ be,
                                  const float* __restrict__ wea,
                                  const float* __restrict__ asrc, const float* __restrict__ adst,
                                  const int* __restrict__ src, const int* __restrict__ dst,
                                  float* __restrict__ alpha, float* __restrict__ m) {
  int e = blockIdx.x * blockDim.x + threadIdx.x;
  if (e >= kE) return;
  const float ea0 = ea[2 * e], ea1 = ea[2 * e + 1];
  float ae[kH] = {0.f, 0.f, 0.f, 0.f};
  #pragma unroll
  for (int j = 0; j < kED; ++j) {
    float t = fmaf(ea0, We[j], fmaf(ea1, We[kED + j], be[j]));
    t = fmaxf(t, 0.f);
    #pragma unroll
    for (int h = 0; h < kH; ++h) ae[h] = fmaf(t, wea[j * kH + h], ae[h]);
  }
  const int s = src[e], d = dst[e];
  #pragma unroll
  for (int h = 0; h < kH; ++h) {
    float a = asrc[s * kH + h] + adst[d * kH + h] + ae[h];
    a = (a > 0.f) ? a : 0.2f * a;            // leaky_relu 0.2
    alpha[e * kH + h] = a;
    atomic_max_f32(&m[d * kH + h], a);
  }
}

__global__ void edge_exp_kernel(float* __restrict__ alpha, const float* __restrict__ m,
                                float* __restrict__ ssum, const int* __restrict__ dst) {
  int e = blockIdx.x * blockDim.x + threadIdx.x;
  if (e >= kE) return;
  const int d = dst[e];
  #pragma unroll
  for (int h = 0; h < kH; ++h) {
    float ex = __expf(alpha[e * kH + h] - m[d * kH + h]);
    alpha[e * kH + h] = ex;
    atomicAdd(&ssum[d * kH + h], ex);
  }
}

__global__ void edge_norm_kernel(float* __restrict__ alpha, const float* __restrict__ ssum,
                                 const int* __restrict__ dst) {
  int i = blockIdx.x * blockDim.x + threadIdx.x;
  if (i >= kE * kH) return;
  int e = i >> 2, h = i & 3;
  int d = dst[e];
  alpha[i] = alpha[i] / (ssum[d * kH + h] + 1e-16f);
}

// acc[d, c] += (1/H) * sum_h proj[s, h*64+c] * w[e, h]   (head-mean folded into scatter)
__global__ void scatter_kernel(const float* __restrict__ proj, const float* __restrict__ w,
                               const int* __restrict__ src, const int* __restrict__ dst,
                               float* __restrict__ acc) {
  int i = blockIdx.x * blockDim.x + threadIdx.x;
  if (i >= kE * kHID) return;                 // 51.2M threads
  int e = i >> 6, c = i & (kHID - 1);
  int s = src[e], d = dst[e];
  const float* __restrict__ xp = proj + (size_t)s * kHH + c;
  const float* __restrict__ wp = w + (size_t)e * kH;
  float msg = xp[0] * wp[0] + xp[64] * wp[1] + xp[128] * wp[2] + xp[192] * wp[3];
  atomicAdd(&acc[(size_t)d * kHID + c], msg * 0.25f);
}

__global__ void finalize_kernel(const float* __restrict__ acc, const float* __restrict__ bias,
                                float* __restrict__ out, int n, int relu) {
  int i = blockIdx.x * blockDim.x + threadIdx.x;
  if (i >= n * kHID) return;
  int c = i & (kHID - 1);
  float v = acc[i] + bias[c];
  if (relu) v = fmaxf(v, 0.f);
  out[i] = v;
}

struct GatBufs {
  float *proj, *asrc, *adst, *alpha, *m, *s, *acc, *wsa, *wda, *wea;
};

static inline int cdiv(int a, int b) { return (a + b - 1) / b; }

static void run_gat(hipStream_t st,
                    const float* x_src, int Nsrc, const float* x_dst, int Ndst,
                    const int* src, const int* dst, const float* ea,
                    const float* We, const float* be,
                    const float* Wsrc, const float* Wdst, const float* Wedge,
                    const float* a_s, const float* a_d, const float* a_e,
                    const float* bias, float* outp, int relu, const GatBufs& B) {
  collapse_kernel<<<cdiv(kHID * kH, 128), 128, 0, st>>>(Wsrc, a_s, B.wsa, kHID);
  collapse_kernel<<<cdiv(kHID * kH, 128), 128, 0, st>>>(Wdst, a_d, B.wda, kHID);
  collapse_kernel<<<cdiv(kED * kH, 128), 128, 0, st>>>(Wedge, a_e, B.wea, kED);

  proj_wmma_kernel<<<Nsrc / 16, 128, 0, st>>>(x_src, Wsrc, B.proj);

  node_alpha_kernel<<<cdiv(Nsrc * kH, 256), 256, 0, st>>>(x_src, B.wsa, B.asrc, Nsrc);
  node_alpha_kernel<<<cdiv(Ndst * kH, 256), 256, 0, st>>>(x_dst, B.wda, B.adst, Ndst);

  fill_kernel<<<cdiv(Ndst * kH, 256), 256, 0, st>>>(B.m, -__builtin_huge_valf(), Ndst * kH);
  fill_kernel<<<cdiv(Ndst * kH, 256), 256, 0, st>>>(B.s, 0.f, Ndst * kH);
  fill_kernel<<<cdiv(Ndst * kHID, 256), 256, 0, st>>>(B.acc, 0.f, Ndst * kHID);

  edge_alpha_kernel<<<cdiv(kE, 256), 256, 0, st>>>(ea, We, be, B.wea, B.asrc, B.adst,
                                                   src, dst, B.alpha, B.m);
  edge_exp_kernel<<<cdiv(kE, 256), 256, 0, st>>>(B.alpha, B.m, B.s, dst);
  edge_norm_kernel<<<cdiv(kE * kH, 256), 256, 0, st>>>(B.alpha, B.s, dst);
  scatter_kernel<<<cdiv(kE * kHID, 256), 256, 0, st>>>(B.proj, B.alpha, src, dst, B.acc);
  finalize_kernel<<<cdiv(Ndst * kHID, 256), 256, 0, st>>>(B.acc, bias, outp, Ndst, relu);
}

} // namespace

extern "C" void kernel_launch(void* const* d_in, const int* in_sizes, int n_in,
                              void* d_out, int out_size, void* d_ws, size_t ws_size,
                              hipStream_t stream) {
  (void)in_sizes; (void)n_in; (void)out_size; (void)ws_size;

  const int* x_student = (const int*)d_in[0];
  const int* x_course  = (const int*)d_in[1];
  const int* src_takes = (const int*)d_in[2];
  const int* dst_takes = (const int*)d_in[3];
  const int* src_rev   = (const int*)d_in[4];
  const int* dst_rev   = (const int*)d_in[5];
  const float* ea_takes = (const float*)d_in[6];
  const float* ea_rev   = (const float*)d_in[7];
  const float* emb_student = (const float*)d_in[8];
  const float* emb_course  = (const float*)d_in[9];
  const float* We_takes = (const float*)d_in[10];
  const float* be_takes = (const float*)d_in[11];
  const float* We_rev   = (const float*)d_in[12];
  const float* be_rev   = (const float*)d_in[13];
  const float* Wsrc_t  = (const float*)d_in[14];
  const float* Wdst_t  = (const float*)d_in[15];
  const float* Wedge_t = (const float*)d_in[16];
  const float* asrc_t  = (const float*)d_in[17];
  const float* adst_t  = (const float*)d_in[18];
  const float* aedge_t = (const float*)d_in[19];
  const float* bias_t  = (const float*)d_in[20];
  const float* Wsrc_r  = (const float*)d_in[21];
  const float* Wdst_r  = (const float*)d_in[22];
  const float* Wedge_r = (const float*)d_in[23];
  const float* asrc_r  = (const float*)d_in[24];
  const float* adst_r  = (const float*)d_in[25];
  const float* aedge_r = (const float*)d_in[26];
  const float* bias_r  = (const float*)d_in[27];

  float* ws = (float*)d_ws;
  size_t off = 0;
  auto take = [&](size_t n) { float* p = ws + off; off += n; return p; };

  float* hsA = take((size_t)kNS * kHID);
  float* hsB = take((size_t)kNS * kHID);
  float* hcA = take((size_t)kNC * kHID);
  float* hcB = take((size_t)kNC * kHID);

  GatBufs B;
  B.proj  = take((size_t)kNS * kHH);
  B.asrc  = take((size_t)kNS * kH);
  B.adst  = take((size_t)kNS * kH);
  B.alpha = take((size_t)kE * kH);
  B.m     = take((size_t)kNS * kH);
  B.s     = take((size_t)kNS * kH);
  B.acc   = take((size_t)kNS * kHID);
  B.wsa   = take((size_t)kHID * kH);
  B.wda   = take((size_t)kHID * kH);
  B.wea   = take((size_t)kED * kH);

  gather_kernel<<<cdiv(kNS * kHID, 256), 256, 0, stream>>>(emb_student, x_student, hsA, kNS);
  gather_kernel<<<cdiv(kNC * kHID, 256), 256, 0, stream>>>(emb_course, x_course, hcA, kNC);

  float* out_hs = (float*)d_out;
  float* out_hc = (float*)d_out + (size_t)kNS * kHID;

  float *hs_cur = hsA, *hs_nxt = hsB, *hc_cur = hcA, *hc_nxt = hcB;
  for (int l = 0; l < kL; ++l) {
    const int last = (l == kL - 1);
    float* outc = last ? out_hc : hc_nxt;
    float* outs = last ? out_hs : hs_nxt;

    // takes: student -> course
    run_gat(stream, hs_cur, kNS, hc_cur, kNC, src_takes, dst_takes, ea_takes,
            We_takes, be_takes,
            Wsrc_t + (size_t)l * kHID * kHH, Wdst_t + (size_t)l * kHID * kHH,
            Wedge_t + (size_t)l * kED * kHH,
            asrc_t + (size_t)l * kH * kHID, adst_t + (size_t)l * kH * kHID,
            aedge_t + (size_t)l * kH * kHID,
            bias_t + (size_t)l * kHID, outc, !last, B);

    // rev: course -> student (reads OLD hs_cur/hc_cur; new_c went to outc, not hc_cur)
    run_gat(stream, hc_cur, kNC, hs_cur, kNS, src_rev, dst_rev, ea_rev,
            We_rev, be_rev,
            Wsrc_r + (size_t)l * kHID * kHH, Wdst_r + (size_t)l * kHID * kHH,
            Wedge_r + (size_t)l * kED * kHH,
            asrc_r + (size_t)l * kH * kHID, adst_r + (size_t)l * kH * kHID,
            aedge_r + (size_t)l * kH * kHID,
            bias_r + (size_t)l * kHID, outs, !last, B);

    float* t;
    t = hs_cur; hs_cur = hs_nxt; hs_nxt = t;
    t = hc_cur; hc_cur = hc_nxt; hc_nxt = t;
  }
}